// PureGraphEncoder_36206574306115
// MI455X (gfx1250) — compile-verified
//
#include <hip/hip_runtime.h>
#include <hip/hip_bf16.h>

typedef __attribute__((ext_vector_type(2))) float v2f;
typedef __attribute__((ext_vector_type(8))) float v8f;

#define GCN_N 100000
#define GCN_E 1600000
#define GCN_DIM 128
#define GCN_MASK 5000

// Hardware f32 atomic add at device scope (L2 is the coherence point; non-returning
// form uses STOREcnt and S_ENDPGM's implicit wait-idle flushes it).
__device__ __forceinline__ void atomic_add_f32_dev(float* p, float v) {
    asm volatile("global_atomic_add_f32 %0, %1, off scope:SCOPE_DEV"
                 :: "v"(p), "v"(v) : "memory");
}

// ---------------- degree / normalization ----------------

__global__ void gcn_init_deg(float* __restrict__ deg, int n) {
    int i = blockIdx.x * blockDim.x + threadIdx.x;
    if (i < n) deg[i] = 1.0f;   // self-loop weight
}

__global__ void gcn_deg_scatter(const int* __restrict__ dst,
                                const float* __restrict__ w,
                                float* __restrict__ deg, int e) {
    int i = blockIdx.x * blockDim.x + threadIdx.x;
    if (i < e) atomic_add_f32_dev(&deg[dst[i]], w[i]);
}

__global__ void gcn_dis(const float* __restrict__ deg,
                        float* __restrict__ dis, int n) {
    int i = blockIdx.x * blockDim.x + threadIdx.x;
    if (i < n) {
        float d = deg[i];
        dis[i] = (d > 0.0f) ? rsqrtf(d) : 0.0f;
    }
}

// ---------------- GEMM: H = (relu?)X @ W via V_WMMA_F32_16X16X4_F32 ----------------
// Block = 128 threads = 4 waves. Block computes a 16x128 strip of H.
// Each wave owns two 16x16 N-tiles. K loop: 128/4 = 32 WMMA steps per tile.

template <bool RELU>
__global__ __launch_bounds__(128)
void gcn_gemm_wmma(const float* __restrict__ X,
                   const float* __restrict__ W,
                   float* __restrict__ H) {
    __shared__ float tile[16][132];   // 132: pad so A-frag column reads hit distinct banks

    const int tid  = threadIdx.x;
    const int wave = tid >> 5;
    const int lane = tid & 31;
    const int m0   = blockIdx.x * 16;

    // Stage the 16x128 X tile into LDS with coalesced float4 loads.
    {
        const float4* src = (const float4*)(X + (size_t)m0 * GCN_DIM);
        for (int p = tid; p < 512; p += 128) {      // 512 float4 = 16x128 floats
            int r = p >> 5, c = p & 31;
            float4 v = src[(size_t)r * 32 + c];
            if (RELU) {
                v.x = fmaxf(v.x, 0.0f); v.y = fmaxf(v.y, 0.0f);
                v.z = fmaxf(v.z, 0.0f); v.w = fmaxf(v.w, 0.0f);
            }
            tile[r][c * 4 + 0] = v.x; tile[r][c * 4 + 1] = v.y;
            tile[r][c * 4 + 2] = v.z; tile[r][c * 4 + 3] = v.w;
        }
    }
    __syncthreads();

    const int mr   = lane & 15;
    const int ksel = (lane < 16) ? 0 : 2;          // A/B frag: lanes 16-31 hold K+2,K+3
    const int col0 = wave * 32 + mr;               // first N-tile column for this lane

    v8f acc0 = {};
    v8f acc1 = {};

#pragma unroll 4
    for (int k = 0; k < GCN_DIM; k += 4) {
        v2f a;
        a.x = tile[mr][k + ksel];
        a.y = tile[mr][k + ksel + 1];

        const float* Wk = W + (size_t)(k + ksel) * GCN_DIM;
        v2f b0, b1;
        b0.x = Wk[col0];            b0.y = Wk[GCN_DIM + col0];
        b1.x = Wk[col0 + 16];       b1.y = Wk[GCN_DIM + col0 + 16];

        acc0 = __builtin_amdgcn_wmma_f32_16x16x4_f32(false, a, false, b0,
                                                     (short)0, acc0, false, false);
        acc1 = __builtin_amdgcn_wmma_f32_16x16x4_f32(false, a, false, b1,
                                                     (short)0, acc1, false, false);
    }

    // C/D layout: VGPR r -> M = r + (lane<16 ? 0 : 8), N = lane&15
    const int rowBase = m0 + ((lane < 16) ? 0 : 8);
    const int n0      = wave * 32;
#pragma unroll
    for (int r = 0; r < 8; ++r) {
        H[(size_t)(rowBase + r) * GCN_DIM + n0 + mr]      = acc0[r];
        H[(size_t)(rowBase + r) * GCN_DIM + n0 + 16 + mr] = acc1[r];
    }
}

// ---------------- aggregation ----------------
// AGG[n][:] = dis[n]^2 * H[n][:] + b[:]   (self loop + bias)

__global__ void gcn_agg_init(const float* __restrict__ H,
                             const float* __restrict__ dis,
                             const float* __restrict__ bias,
                             float* __restrict__ AGG, int n) {
    int idx = blockIdx.x * blockDim.x + threadIdx.x;   // n * 32 float4 slots
    int node = idx >> 5, c4 = idx & 31;
    if (node >= n) return;
    float s = dis[node]; s *= s;
    float4 h = ((const float4*)(H + (size_t)node * GCN_DIM))[c4];
    float4 b = ((const float4*)bias)[c4];
    float4 o = { s * h.x + b.x, s * h.y + b.y, s * h.z + b.z, s * h.w + b.w };
    ((float4*)(AGG + (size_t)node * GCN_DIM))[c4] = o;
}

// AGG[dst][:] += dis[src]*w*dis[dst] * H[src][:]  — one wave per edge, L2-resident atomics

__global__ __launch_bounds__(256)
void gcn_edge_scatter(const float* __restrict__ H,
                      float* __restrict__ AGG,
                      const int* __restrict__ src,
                      const int* __restrict__ dst,
                      const float* __restrict__ w,
                      const float* __restrict__ dis, int e) {
    int gw   = (blockIdx.x * blockDim.x + threadIdx.x) >> 5;  // wave32: one edge per wave
    int lane = threadIdx.x & 31;
    if (gw >= e) return;
    int s = src[gw], d = dst[gw];
    float coef = dis[s] * w[gw] * dis[d];
    float4 v = ((const float4*)(H + (size_t)s * GCN_DIM))[lane];  // 32 lanes x 16B = 512B
    float* out = AGG + (size_t)d * GCN_DIM + lane * 4;
    atomic_add_f32_dev(out + 0, coef * v.x);
    atomic_add_f32_dev(out + 1, coef * v.y);
    atomic_add_f32_dev(out + 2, coef * v.z);
    atomic_add_f32_dev(out + 3, coef * v.w);
}

// ---------------- gather outputs ----------------

__global__ void gcn_gather_h(const float* __restrict__ AGG,
                             const int* __restrict__ mask,
                             float* __restrict__ out) {
    int idx = blockIdx.x * blockDim.x + threadIdx.x;   // GCN_MASK * 32 float4 slots
    int i = idx >> 5, c4 = idx & 31;
    if (i >= GCN_MASK) return;
    int n = mask[i];
    ((float4*)(out + (size_t)i * GCN_DIM))[c4] =
        ((const float4*)(AGG + (size_t)n * GCN_DIM))[c4];
}

__global__ void gcn_gather_y(const int* __restrict__ y,
                             const int* __restrict__ mask,
                             float* __restrict__ out) {
    int i = blockIdx.x * blockDim.x + threadIdx.x;
    if (i < GCN_MASK) out[i] = (float)y[mask[i]];
}

// ---------------- launcher ----------------

extern "C" void kernel_launch(void* const* d_in, const int* in_sizes, int n_in,
                              void* d_out, int out_size, void* d_ws, size_t ws_size,
                              hipStream_t stream) {
    const float* x    = (const float*)d_in[0];
    const float* ew   = (const float*)d_in[1];
    const float* W1   = (const float*)d_in[2];
    const float* b1   = (const float*)d_in[3];
    const float* W2   = (const float*)d_in[4];
    const float* b2   = (const float*)d_in[5];
    const int*   eidx = (const int*)d_in[6];
    const int*   mask = (const int*)d_in[7];
    const int*   y    = (const int*)d_in[8];
    const int* src = eidx;
    const int* dst = eidx + GCN_E;

    float* deg  = (float*)d_ws;                 // N
    float* dis  = deg + GCN_N;                  // N
    float* bufH = dis + GCN_N;                  // N*128 (GEMM output)
    float* bufA = bufH + (size_t)GCN_N * GCN_DIM; // N*128 (aggregated)

    float* out_h = (float*)d_out;               // 5000*128
    float* out_y = out_h + (size_t)GCN_MASK * GCN_DIM; // 5000

    const int B = 256;
    // normalization
    gcn_init_deg<<<(GCN_N + B - 1) / B, B, 0, stream>>>(deg, GCN_N);
    gcn_deg_scatter<<<(GCN_E + B - 1) / B, B, 0, stream>>>(dst, ew, deg, GCN_E);
    gcn_dis<<<(GCN_N + B - 1) / B, B, 0, stream>>>(deg, dis, GCN_N);

    const int gemmBlocks = GCN_N / 16;                 // 6250, exact
    const int aggThreads = GCN_N * 32;                 // float4 slots
    const int edgThreads = GCN_E * 32;                 // wave per edge

    // layer 1
    gcn_gemm_wmma<false><<<gemmBlocks, 128, 0, stream>>>(x, W1, bufH);
    gcn_agg_init<<<(aggThreads + B - 1) / B, B, 0, stream>>>(bufH, dis, b1, bufA, GCN_N);
    gcn_edge_scatter<<<(edgThreads + B - 1) / B, B, 0, stream>>>(bufH, bufA, src, dst, ew, dis, GCN_E);

    // layer 2 (ReLU folded into GEMM input staging)
    gcn_gemm_wmma<true><<<gemmBlocks, 128, 0, stream>>>(bufA, W2, bufH);
    gcn_agg_init<<<(aggThreads + B - 1) / B, B, 0, stream>>>(bufH, dis, b2, bufA, GCN_N);
    gcn_edge_scatter<<<(edgThreads + B - 1) / B, B, 0, stream>>>(bufH, bufA, src, dst, ew, dis, GCN_E);

    // outputs
    gcn_gather_h<<<(GCN_MASK * 32 + B - 1) / B, B, 0, stream>>>(bufA, mask, out_h);
    gcn_gather_y<<<(GCN_MASK + B - 1) / B, B, 0, stream>>>(y, mask, out_y);
}